// Decoder_51831665328592
// MI455X (gfx1250) — compile-verified
//
#include <hip/hip_runtime.h>
#include <hip/hip_bf16.h>
#include <math.h>

// ---------------------------------------------------------------------------
// Problem constants (from reference)
// ---------------------------------------------------------------------------
#define NUM_PLACES 81
#define H_DEC      100
#define FOURH      400     // 4 * H_DEC
#define H_ENC      512
#define BATCHSZ    4096

#define TILE_B     16      // batch rows per workgroup (one WMMA M tile)
#define NWAVES     5       // 5 waves * 5 column tiles = 25 tiles = 400 cols
#define THREADS    (NWAVES * 32)
#define NTILE      25      // 400 / 16 column tiles
#define KT_X       128     // 512 / 4 K-steps for x@W_x
#define KT_H       25      // 100 / 4 K-steps for h@W_h
#define KT_TOT     (KT_X + KT_H)

typedef float v2f __attribute__((ext_vector_type(2)));
typedef float v8f __attribute__((ext_vector_type(8)));

// gfx1250 async global->LDS path (ASYNCcnt), with safe fallback
#if defined(__AMDGCN__) && __has_builtin(__builtin_amdgcn_global_load_async_to_lds_b128)
#define USE_ASYNC_LDS 1
#define GLOBAL_AS __attribute__((address_space(1)))
#define LDS_AS    __attribute__((address_space(3)))
// builtin pointer params are: <16-byte int vector> addrspace-qualified pointers
typedef int v4i_vs __attribute__((vector_size(16)));
#else
#define USE_ASYNC_LDS 0
#endif

__device__ __forceinline__ float sigmoidf_(float x) {
    return 1.0f / (1.0f + __expf(-x));
}

__device__ __forceinline__ void wait_async_lds() {
#if USE_ASYNC_LDS
#if __has_builtin(__builtin_amdgcn_s_wait_asynccnt)
    __builtin_amdgcn_s_wait_asynccnt(0);
#else
    asm volatile("s_wait_asynccnt 0x0" ::: "memory");
#endif
#endif
}

// Stage the 16x512 f32 x-tile for step p into LDS buf.
// Async variant: fire-and-forget; consumer must wait_async_lds() + barrier.
__device__ __forceinline__ void stage_x(const float* __restrict__ h_enc,
                                        float* buf, int b0, int p, int tid) {
    for (int idx = tid; idx < TILE_B * H_ENC / 4; idx += THREADS) {
        int row = idx >> 7;          // / (512/4)
        int e4  = idx & 127;
        size_t goff = ((size_t)(b0 + row) * NUM_PLACES + p) * H_ENC + (size_t)e4 * 4;
#if USE_ASYNC_LDS
        __builtin_amdgcn_global_load_async_to_lds_b128(
            (GLOBAL_AS v4i_vs*)(h_enc + goff),
            (LDS_AS v4i_vs*)(buf + row * H_ENC + e4 * 4),
            0 /*offset*/, 0 /*cpol*/);
#else
        ((float4*)buf)[row * 128 + e4] = *(const float4*)(h_enc + goff);
#endif
    }
}

// ---------------------------------------------------------------------------
// Kernel 0: pack [W_x ; W_h] (612 x 400, row-major) into WMMA B-fragment
// order.  Fragment (t, u) covers K rows 4t..4t+3 and N cols 16u..16u+15.
// f32 16x16x4 layout: lane L (L<16)  holds K = 4t+0,4t+1 for N=16u+L,
//                     lane L (L>=16) holds K = 4t+2,4t+3 for N=16u+L-16.
// float2 per lane: frag[((t*25 + u)*32 + lane)*2 + {0,1}]   (~0.98 MB in d_ws)
// ---------------------------------------------------------------------------
__global__ void pack_weights_kernel(const float* __restrict__ Wx,
                                    const float* __restrict__ Wh,
                                    float* __restrict__ frag) {
    const int TOTAL = KT_TOT * NTILE * 32;
    int idx = blockIdx.x * blockDim.x + threadIdx.x;
    if (idx >= TOTAL) return;

    int lane = idx & 31;
    int u    = (idx >> 5) % NTILE;
    int t    = idx / (32 * NTILE);

    int kbase = 4 * t + 2 * (lane >> 4);
    int n     = u * 16 + (lane & 15);

    int k0 = kbase, k1 = kbase + 1;
    float v0 = (k0 < H_ENC) ? Wx[(size_t)k0 * FOURH + n]
                            : Wh[(size_t)(k0 - H_ENC) * FOURH + n];
    float v1 = (k1 < H_ENC) ? Wx[(size_t)k1 * FOURH + n]
                            : Wh[(size_t)(k1 - H_ENC) * FOURH + n];

    v2f out; out.x = v0; out.y = v1;
    *(v2f*)(frag + 2 * (size_t)idx) = out;
}

// ---------------------------------------------------------------------------
// Kernel 1: fused LSTM scan + masked softmax.
// One workgroup = 16 batch rows, all 81 steps, h/c state kept on-chip.
// Software pipeline: x-tile for step p+1 streams into LDS (ASYNCcnt) while
// the softmax reductions and output stores for step p execute.
// ---------------------------------------------------------------------------
__global__ __launch_bounds__(THREADS)
void decoder_fused_kernel(const float* __restrict__ h_enc,   // [B, P, E]
                          const float* __restrict__ h0,      // [B, H]
                          const float* __restrict__ bias,    // [4H]
                          const int*   __restrict__ mask,    // [P, H]
                          const float* __restrict__ frag,    // packed weights
                          float*       __restrict__ out) {   // [B, P, H]
    __shared__ float buf[TILE_B * H_ENC];     // x tile (async-filled)
    __shared__ float zbuf[TILE_B * FOURH];    // gate pre-activations z
    __shared__ float h_lds[TILE_B * H_DEC];   // recurrent state h
    __shared__ float red[THREADS];            // partial reductions
    __shared__ float rowred[TILE_B];          // per-row max / sum
    // total LDS: 32768 + 25600 + 6400 + 640 + 64 = 65472 B

    const int tid  = threadIdx.x;
    const int lane = tid & 31;
    const int wave = tid >> 5;
    const int b0   = blockIdx.x * TILE_B;

    const int m16 = lane & 15;   // WMMA A row / B column lane
    const int hl  = lane >> 4;   // half-wave select

    // gate-phase ownership: thread -> (row gm, 10 columns starting at jbase)
    const int gm    = tid & 15;
    const int jbase = (tid >> 4) * 10;

    float creg[10];              // private cell state c
#pragma unroll
    for (int j = 0; j < 10; ++j) creg[j] = 0.0f;

    // kick off x-tile for step 0, init h from h0
    stage_x(h_enc, buf, b0, 0, tid);
#pragma unroll
    for (int j = 0; j < 10; ++j)
        h_lds[gm * H_DEC + jbase + j] = h0[(size_t)(b0 + gm) * H_DEC + jbase + j];

    for (int p = 0; p < NUM_PLACES; ++p) {
        wait_async_lds();
        __syncthreads();   // x(p) staged, h(p) visible

        // ---- z tile accumulators: 5 column tiles per wave ----
        v8f acc[5];
#pragma unroll
        for (int j = 0; j < 5; ++j)
#pragma unroll
            for (int r = 0; r < 8; ++r) acc[j][r] = 0.0f;

        // x @ W_x  (K = 512)
        for (int t = 0; t < KT_X; ++t) {
            v2f a = *(const v2f*)(buf + m16 * H_ENC + 4 * t + 2 * hl);
            const float* fb = frag + ((size_t)t * NTILE + wave * 5) * 64 + lane * 2;
            __builtin_prefetch(fb + (size_t)NTILE * 64, 0, 1);  // next K-step
#pragma unroll
            for (int j = 0; j < 5; ++j) {
                v2f bf = *(const v2f*)(fb + j * 64);
                acc[j] = __builtin_amdgcn_wmma_f32_16x16x4_f32(
                    false, a, false, bf, (short)0, acc[j], false, false);
            }
        }
        // h @ W_h  (K = 100)
        for (int t = 0; t < KT_H; ++t) {
            v2f a = *(const v2f*)(h_lds + m16 * H_DEC + 4 * t + 2 * hl);
            const float* fb = frag + ((size_t)(KT_X + t) * NTILE + wave * 5) * 64 + lane * 2;
#pragma unroll
            for (int j = 0; j < 5; ++j) {
                v2f bf = *(const v2f*)(fb + j * 64);
                acc[j] = __builtin_amdgcn_wmma_f32_16x16x4_f32(
                    false, a, false, bf, (short)0, acc[j], false, false);
            }
        }

        // ---- store z (+bias) into zbuf[16][400] ----
        // C/D layout: VGPR r, lanes 0-15 -> M=r, lanes 16-31 -> M=r+8; N=lane%16
#pragma unroll
        for (int j = 0; j < 5; ++j) {
            int n = (wave * 5 + j) * 16 + m16;
            float bn = bias[n];
#pragma unroll
            for (int r = 0; r < 8; ++r)
                zbuf[(r + 8 * hl) * FOURH + n] = acc[j][r] + bn;
        }
        __syncthreads();   // z complete; all waves past WMMA reads of buf

        // ---- gates: i,f,g,o = z[:,0:100],[100:200],[200:300],[300:400] ----
        float lg[10];
#pragma unroll
        for (int j = 0; j < 10; ++j) {
            int col = jbase + j;
            float zi = zbuf[gm * FOURH + col];
            float zf = zbuf[gm * FOURH + col + H_DEC];
            float zg = zbuf[gm * FOURH + col + 2 * H_DEC];
            float zo = zbuf[gm * FOURH + col + 3 * H_DEC];
            float c  = sigmoidf_(zf) * creg[j] + sigmoidf_(zi) * zg;
            creg[j] = c;
            float hn = sigmoidf_(zo) * c;
            h_lds[gm * H_DEC + col] = hn;
            lg[j] = (mask[p * H_DEC + col] != 0) ? hn : -1e9f;
        }

        // ---- overlap: start streaming next step's x-tile into buf ----
        if (p + 1 < NUM_PLACES)
            stage_x(h_enc, buf, b0, p + 1, tid);

        // ---- masked softmax over the 100 columns of each row ----
        float mx = lg[0];
#pragma unroll
        for (int j = 1; j < 10; ++j) mx = fmaxf(mx, lg[j]);
        red[tid] = mx;
        __syncthreads();
        if (tid < TILE_B) {
            float m2 = red[tid];
#pragma unroll
            for (int c = 1; c < 10; ++c) m2 = fmaxf(m2, red[tid + 16 * c]);
            rowred[tid] = m2;
        }
        __syncthreads();
        float rm = rowred[gm];
        float ex[10];
        float s = 0.0f;
#pragma unroll
        for (int j = 0; j < 10; ++j) { ex[j] = __expf(lg[j] - rm); s += ex[j]; }
        red[tid] = s;
        __syncthreads();    // also retires all reads of rowred (rm)
        if (tid < TILE_B) {
            float s2 = 0.0f;
#pragma unroll
            for (int c = 0; c < 10; ++c) s2 += red[tid + 16 * c];
            rowred[tid] = s2;
        }
        __syncthreads();
        float rinv = 1.0f / rowred[gm];
        float* orow = out + ((size_t)(b0 + gm) * NUM_PLACES + p) * H_DEC + jbase;
#pragma unroll
        for (int j = 0; j < 10; ++j) orow[j] = ex[j] * rinv;
        // next loop-top wait+barrier orders everything for step p+1
    }
}

// ---------------------------------------------------------------------------
// Launch
// ---------------------------------------------------------------------------
extern "C" void kernel_launch(void* const* d_in, const int* in_sizes, int n_in,
                              void* d_out, int out_size, void* d_ws, size_t ws_size,
                              hipStream_t stream) {
    const float* h_enc = (const float*)d_in[0];   // [4096, 81, 512] f32
    const float* h0    = (const float*)d_in[1];   // [4096, 100]    f32
    const float* W_x   = (const float*)d_in[2];   // [512, 400]     f32
    const float* W_h   = (const float*)d_in[3];   // [100, 400]     f32
    const float* b     = (const float*)d_in[4];   // [400]          f32
    const int*   mask  = (const int*)d_in[5];     // [81, 100]      bool->int
    float*       out   = (float*)d_out;           // [4096, 81, 100] f32
    float*       frag  = (float*)d_ws;            // ~0.98 MB packed weights

    (void)in_sizes; (void)n_in; (void)out_size; (void)ws_size;

    const int TOTAL = KT_TOT * NTILE * 32;        // 122400 fragments
    pack_weights_kernel<<<(TOTAL + 255) / 256, 256, 0, stream>>>(W_x, W_h, frag);

    decoder_fused_kernel<<<BATCHSZ / TILE_B, THREADS, 0, stream>>>(
        h_enc, h0, b, mask, frag, out);
}